// TorchRocket_48318382080036
// MI455X (gfx1250) — compile-verified
//
#include <hip/hip_runtime.h>
#include <hip/hip_bf16.h>

// Problem constants (match reference)
#define Bb     64
#define Cc     12
#define Tt     2048
#define Kk     2048
#define LMAXx  11
#define CMAXx  11
#define PLpad  1024
#define NWAVES 8

// Padded y accumulator: every scatter target PADY + mrow + tau0 - l*d is in-bounds
// for l in [0,15], d <= 341 (= (T-1)/(Lmin-1)), tau0+15 < T + (Leff-1)*d + 31.
#define PADY   5120
#define YTOT   9280

typedef float v2f __attribute__((ext_vector_type(2)));
typedef float v8f __attribute__((ext_vector_type(8)));
typedef unsigned int v4u __attribute__((ext_vector_type(4)));
typedef int v4i __attribute__((ext_vector_type(4)));
typedef int v8i __attribute__((ext_vector_type(8)));

// Issue one TDM 2-D tile load: 12 rows x 2048 cols of f32, global row stride 2048,
// packed contiguously into LDS at lds_off. Tracked by TENSORcnt.
static __device__ __forceinline__ void tdm_load_slab(const float* gsrc, unsigned lds_off) {
    const unsigned long long ga = (unsigned long long)(uintptr_t)gsrc;
    v4u g0;
    g0[0] = 1u;                                          // count=1 (valid user descriptor)
    g0[1] = lds_off;                                     // lds_addr
    g0[2] = (unsigned)(ga & 0xFFFFFFFFu);                // global_addr[31:0]
    g0[3] = (unsigned)((ga >> 32) & 0x01FFFFFFu)         // global_addr[56:32]
          | (2u << 30);                                  // type=2 ("image")
    v8i g1;
    g1[0] = (int)(2u << 16);                             // data_size=2 (4 bytes)
    g1[1] = (int)((Tt & 0xFFFFu) << 16);                 // tensor_dim0[15:0]
    g1[2] = (int)((Tt >> 16) | (Cc << 16));              // tensor_dim0[31:16] | tensor_dim1[15:0]
    g1[3] = (int)((unsigned)Tt << 16);                   // tensor_dim1[31:16]=0 | tile_dim0
    g1[4] = (int)Cc;                                     // tile_dim1 | tile_dim2=0
    g1[5] = (int)Tt;                                     // tensor_dim0_stride[31:0]
    g1[6] = 0;                                           // stride0[47:32] | stride1[15:0]
    g1[7] = 0;                                           // stride1[47:16]
    v4i z4 = {0, 0, 0, 0};
#if __clang_major__ >= 23
    v8i z8 = {0, 0, 0, 0, 0, 0, 0, 0};
    __builtin_amdgcn_tensor_load_to_lds(g0, g1, z4, z4, z8, 0);
#else
    __builtin_amdgcn_tensor_load_to_lds(g0, g1, z4, z4, 0);
#endif
}

__global__ __launch_bounds__(256, 1)
void TorchRocket_48318382080036_kernel(const float* __restrict__ x,
                                       const float* __restrict__ weights,
                                       const float* __restrict__ biases,
                                       const int*   __restrict__ ch_idx,
                                       const int*   __restrict__ dils,
                                       const int*   __restrict__ offs,
                                       const int*   __restrict__ olens,
                                       float*       __restrict__ out)
{
    __shared__ __align__(16) float xs[2][Cc][Tt];        // double-buffered x[b] slab (2 x 96 KB)
    __shared__ __align__(16) float ybuf[YTOT];           // padded shared y accumulator (37 KB)
    __shared__ float Wd[Cc * 16];                        // dense weights [c][l], l padded to 16
    __shared__ int   colnz[16];
    __shared__ float red_ppv[NWAVES], red_mx[NWAVES];

    const int k    = blockIdx.x;
    const int tid  = threadIdx.x;
    const int wave = tid >> 5;
    const int lane = tid & 31;

    const int   d    = dils[k];
    const int   off  = offs[k];
    const int   olen = olens[k];
    const float bias = biases[k];

    // ---- densify weights over channels (zero-padded entries scatter zeros -> exact) ----
    if (tid < Cc * 16) Wd[tid] = 0.0f;
    if (tid < 16)      colnz[tid] = 0;
    __syncthreads();
    if (tid < CMAXx * LMAXx) {
        const int   i = tid / LMAXx, l = tid % LMAXx;
        const int   c = ch_idx[k * CMAXx + i];
        const float w = weights[(size_t)k * (CMAXx * LMAXx) + i * LMAXx + l];
        atomicAdd(&Wd[c * 16 + l], w);
        if (w != 0.0f) colnz[l] = 1;                     // benign race: all writers store 1
    }
    // kick off the TDM load of the first batch slab while weights settle
    if (wave == 0) tdm_load_slab(x, (unsigned)(uintptr_t)&xs[0][0][0]);
    __syncthreads();

    int Leff = 1;                                        // effective kernel length
    for (int l = 0; l < LMAXx; ++l) if (colnz[l]) Leff = l + 1;

    // ---- per-lane A fragments: A[m=l, k=c] = Wd[c][l] for the 3 K=4 channel chunks ----
    // ISA 16x4 f32 A layout: lanes 0-15 hold K-cols {0,1}, lanes 16-31 hold {2,3}; M = lane&15
    const int half = lane >> 4;
    const int mrow = lane & 15;
    v2f afrag[3];
#pragma unroll
    for (int j = 0; j < 3; ++j) {
        const int cA = 4 * j + 2 * half;
        afrag[j].x = Wd[cA * 16 + mrow];
        afrag[j].y = Wd[(cA + 1) * 16 + mrow];
    }

    // per-lane scatter bases: D-row r holds l = r (+8 for upper half); always >= 5
    int sb[8];
#pragma unroll
    for (int r = 0; r < 8; ++r)
        sb[r] = PADY + mrow - (r + 8 * half) * d;

    const int numTiles = (Tt + (Leff - 1) * d + 15) >> 4;
    const int gbase    = off - PLpad;                    // signed offset into un-padded x

    for (int b = 0; b < Bb; ++b) {
        const int p = b & 1;

        // wait for this batch's slab, then prefetch the next one into the other buffer
        if (wave == 0) __builtin_amdgcn_s_wait_tensorcnt(0);
        __syncthreads();
        if (wave == 0 && (b + 1) < Bb)
            tdm_load_slab(x + (size_t)(b + 1) * Cc * Tt,
                          (unsigned)(uintptr_t)&xs[(b + 1) & 1][0][0]);

        // zero the read window of the y accumulator
        for (int t4 = tid; t4 < Tt / 4; t4 += 256)
            ((float4*)(ybuf + PADY))[t4] = make_float4(0.f, 0.f, 0.f, 0.f);
        __syncthreads();

        const float* xcur = &xs[p][0][0];

        // ---- channel-mix GEMM (WMMA) + unguarded diagonal scatter into padded y ----
        for (int tile = wave; tile < numTiles; tile += NWAVES) {
            const int  tau0 = tile << 4;
            const int  g    = gbase + tau0 + mrow;       // time column for this lane
            const bool ok   = (unsigned)g < (unsigned)Tt;
            const int  gc   = ok ? g : 0;                // clamped: branchless ds_load

            v8f acc = {};
#pragma unroll
            for (int j = 0; j < 3; ++j) {
                const int cA = 4 * j + 2 * half;         // B layout mirrors A
                const float r0 = xcur[cA * Tt + gc];
                const float r1 = xcur[(cA + 1) * Tt + gc];
                v2f bfrag;
                bfrag.x = ok ? r0 : 0.0f;
                bfrag.y = ok ? r1 : 0.0f;
                acc = __builtin_amdgcn_wmma_f32_16x16x4_f32(
                        false, afrag[j], false, bfrag, (short)0, acc, false, false);
            }
            // Z[l, tau] -> y[tau - l*d]; out-of-window / zero-weight rows land in pad
#pragma unroll
            for (int r = 0; r < 3; ++r)                  // rows l=0..2 and l=8..10: all lanes
                atomicAdd(&ybuf[sb[r] + tau0], acc[r]);  // ds_add_f32, unguarded
            if (lane < 16) {                             // rows l=11..15 are identically zero
#pragma unroll
                for (int r = 3; r < 8; ++r)              // rows l=3..7: lower half only
                    atomicAdd(&ybuf[sb[r] + tau0], acc[r]);
            }
        }
        __syncthreads();

        // ---- epilogue: PPV + max over valid outputs, whole block cooperates ----
        float ppv = 0.0f;
        float mx  = -INFINITY;
        for (int i = tid; i < Tt / 4; i += 256) {
            const float4 y4 = ((const float4*)(ybuf + PADY))[i];  // ds_load_b128
            const int    t0 = i * 4;
#pragma unroll
            for (int q = 0; q < 4; ++q) {
                const float yv    = (&y4.x)[q] + bias;
                const bool  valid = (t0 + q) < olen;
                const float s     = 1.0f / (1.0f + __expf(3.0f - 7.0f * yv));
                ppv += valid ? s : 0.0f;
                mx   = fmaxf(mx, valid ? yv : -INFINITY);
            }
        }
#pragma unroll
        for (int m = 16; m >= 1; m >>= 1) {              // wave32 shuffle reduction
            ppv += __shfl_xor(ppv, m, 32);
            mx   = fmaxf(mx, __shfl_xor(mx, m, 32));
        }
        if (lane == 0) { red_ppv[wave] = ppv; red_mx[wave] = mx; }
        __syncthreads();
        if (tid == 0) {
            float P = 0.0f, M = -INFINITY;
#pragma unroll
            for (int w = 0; w < NWAVES; ++w) { P += red_ppv[w]; M = fmaxf(M, red_mx[w]); }
            out[(size_t)b * (2 * Kk) + 2 * k]     = P / (float)olen;
            out[(size_t)b * (2 * Kk) + 2 * k + 1] = M;
        }
    }
}

extern "C" void kernel_launch(void* const* d_in, const int* in_sizes, int n_in,
                              void* d_out, int out_size, void* d_ws, size_t ws_size,
                              hipStream_t stream) {
    const float* x       = (const float*)d_in[0];
    const float* weights = (const float*)d_in[1];
    const float* biases  = (const float*)d_in[2];
    const int*   ch_idx  = (const int*)d_in[3];
    const int*   dils    = (const int*)d_in[4];
    const int*   offs    = (const int*)d_in[5];
    const int*   olens   = (const int*)d_in[6];
    float*       out     = (float*)d_out;

    TorchRocket_48318382080036_kernel<<<Kk, 256, 0, stream>>>(
        x, weights, biases, ch_idx, dils, offs, olens, out);
}